// gradient_attention_mean_17712445129432
// MI455X (gfx1250) — compile-verified
//
#include <hip/hip_runtime.h>
#include <hip/hip_bf16.h>
#include <math.h>

typedef float v2f __attribute__((ext_vector_type(2)));
typedef float v8f __attribute__((ext_vector_type(8)));
typedef float f4  __attribute__((ext_vector_type(4), aligned(4)));

#define NP 8
#define NG 16

// ---------------- stage 0: zero Gram scratch (graph-replay safe) ----------------
__global__ void zero_ws_kernel(float* __restrict__ p, int n) {
    int i = blockIdx.x * blockDim.x + threadIdx.x;
    if (i < n) p[i] = 0.0f;
}

// One 8-column Gram step: half-wave exchange + two V_WMMA_F32_16X16X4_F32.
// f holds cols [k,k+4) on lanes 0-15 and [k+4,k+8) on lanes 16-31 of row lane%16.
// A and B operands share registers (B = A^T has the identical VGPR layout).
__device__ __forceinline__ v8f gram_step(f4 f, int hi, v8f acc) {
    float sx = hi ? f.x : f.z;          // lo lanes send (z,w); hi lanes send (x,y)
    float sy = hi ? f.y : f.w;
    float rx = __shfl_xor(sx, 16);
    float ry = __shfl_xor(sy, 16);
    v2f a1, a2;
    a1.x = hi ? rx  : f.x;  a1.y = hi ? ry  : f.y;   // K-block [k,   k+4)
    a2.x = hi ? f.z : rx;   a2.y = hi ? f.w : ry;    // K-block [k+4, k+8)
    acc = __builtin_amdgcn_wmma_f32_16x16x4_f32(false, a1, false, a1, (short)0, acc, false, false);
    acc = __builtin_amdgcn_wmma_f32_16x16x4_f32(false, a2, false, a2, (short)0, acc, false, false);
    return acc;
}

// ---------------- stage 1: Gram matrices via WMMA ----------------
__global__ __launch_bounds__(256) void gram_wmma_kernel(
    const float* __restrict__ g, float* __restrict__ gram,
    long long D, long long chunkK)
{
    const int p    = blockIdx.y;
    const int lane = threadIdx.x & 31;
    const int row  = lane & 15;   // A layout: M = lane%16
    const int hi   = lane >> 4;   // lane half selects K-pair
    const long long wave = (long long)blockIdx.x * (blockDim.x >> 5) + (threadIdx.x >> 5);

    const long long kBeg = wave * chunkK;
    long long kEnd = kBeg + chunkK;
    if (kEnd > D) kEnd = D;

    const float* rp = g + ((size_t)p * NG + row) * (size_t)D;

    // Wave-uniform trip count, scalarized so the loop uses s_cmp/s_cbranch.
    long long span = kEnd - kBeg;
    int nFull = (span > 0) ? (int)(span >> 3) : 0;   // number of 8-col blocks
    nFull = __builtin_amdgcn_readfirstlane(nFull);

    v8f acc = {0.f, 0.f, 0.f, 0.f, 0.f, 0.f, 0.f, 0.f};

    // Depth-2 software pipeline over 8-col blocks: while WMMA-processing blocks
    // (it-2, it-1) the two b128 loads for blocks (it, it+1) are in flight, so
    // the wave keeps 2 loads outstanding (s_wait_loadcnt <= 2, not 0).
    const float* ptr = rp + kBeg + 4 * hi;
    if (nFull >= 2) {
        f4 f0 = *(const f4*)ptr;
        f4 f1 = *(const f4*)(ptr + 8);
        for (int it = 2; it + 2 <= nFull; it += 2) {
            f4 n0 = *(const f4*)(ptr + 16);
            f4 n1 = *(const f4*)(ptr + 24);
            acc = gram_step(f0, hi, acc);
            acc = gram_step(f1, hi, acc);
            ptr += 16;
            f0 = n0;
            f1 = n1;
        }
        acc = gram_step(f0, hi, acc);
        acc = gram_step(f1, hi, acc);
        ptr += 16;
    }
    if (nFull & 1) {                    // odd trailing 8-col block
        f4 f = *(const f4*)ptr;
        acc = gram_step(f, hi, acc);
    }

    // Tail (<8 cols): zero-padded columns contribute nothing to the Gram sum.
    long long k = kBeg + ((long long)nFull << 3);
    if (k < kEnd) {
        long long c = k + 4 * hi;
        f4 f;
        f.x = (c + 0 < kEnd) ? rp[c + 0] : 0.f;
        f.y = (c + 1 < kEnd) ? rp[c + 1] : 0.f;
        f.z = (c + 2 < kEnd) ? rp[c + 2] : 0.f;
        f.w = (c + 3 < kEnd) ? rp[c + 3] : 0.f;
        acc = gram_step(f, hi, acc);
    }

    // Block-level reduction in LDS (ds_add_f32), then one global atomic/thread.
    // C/D layout: VGPR v holds M = v + 8*hi, N = lane%16 -> cell M*16+N.
    __shared__ float red[NG * NG];              // blockDim.x == 256 == NG*NG
    red[threadIdx.x] = 0.0f;
    __syncthreads();
    const int cellBase = hi * 128 + row;        // (8*hi)*16 + row
    #pragma unroll
    for (int v = 0; v < 8; ++v) {
        atomicAdd(&red[v * 16 + cellBase], acc[v]);
    }
    __syncthreads();
    float* gp = gram + (size_t)p * (NG * NG);
    atomicAdd(gp + threadIdx.x, red[threadIdx.x]);
}

// ---------------- stage 2: per-parameter weight solve (tiny) ----------------
__global__ void weights_kernel(const float* __restrict__ gram, float* __restrict__ w) {
    int p = blockIdx.x;
    if (threadIdx.x != 0) return;
    const float* Gm = gram + (size_t)p * (NG * NG);

    float n[NG], rsum[NG], rraw[NG];
    #pragma unroll
    for (int i = 0; i < NG; ++i) n[i] = sqrtf(Gm[i * NG + i]);

    float S = 0.f;
    for (int i = 0; i < NG; ++i) {
        float rw = 0.f, rs = 0.f;
        for (int j = 0; j < NG; ++j) {
            float gij = Gm[i * NG + j];
            rw += gij;
            rs += gij / (n[i] * n[j]);     // cosine row sum (diagonal ~ 1)
        }
        rraw[i] = rw;
        rsum[i] = rs;
        S += rw;                            // S = ||sum_i g_i||^2 >= 0
    }
    float nm = sqrtf(S) / (float)NG;        // ||mean_g||

    bool pos[NG];
    int npos = 0;
    for (int i = 0; i < NG; ++i) {
        float msim = (rraw[i] / (float)NG) / (nm * n[i]);
        pos[i] = msim > 0.f;                // strict, ties -> negative group
        npos += pos[i] ? 1 : 0;
    }
    bool use_pos = npos > (NG - npos);      // strict majority

    float logit[NG];
    float mx = -INFINITY;
    for (int i = 0; i < NG; ++i) {
        bool m = use_pos ? pos[i] : !pos[i];
        logit[i] = m ? rsum[i] : -INFINITY;
        if (logit[i] > mx) mx = logit[i];
    }
    float den = 0.f;
    for (int i = 0; i < NG; ++i) {
        logit[i] = expf(logit[i] - mx);     // exp(-inf - mx) = 0 for masked
        den += logit[i];
    }
    for (int i = 0; i < NG; ++i) w[p * NG + i] = logit[i] / den;
}

// ---------------- stage 3: out[p] = w[p] @ g[p]  (streaming, float4) ----------------
__global__ __launch_bounds__(256) void wsum_kernel(
    const float* __restrict__ g, const float* __restrict__ w,
    float* __restrict__ out, long long D)
{
    const int p = blockIdx.y;
    __shared__ float ws[NG];
    if (threadIdx.x < NG) ws[threadIdx.x] = w[p * NG + threadIdx.x];
    __syncthreads();

    const float* base = g + (size_t)p * NG * (size_t)D;
    float* ob = out + (size_t)p * (size_t)D;
    const long long stride = (long long)gridDim.x * blockDim.x;

    for (long long q = (long long)blockIdx.x * blockDim.x + threadIdx.x;
         q * 4 < D; q += stride) {
        long long d = q * 4;
        if (d + 4 <= D) {
            f4 acc = {0.f, 0.f, 0.f, 0.f};
            #pragma unroll
            for (int i = 0; i < NG; ++i) {
                f4 v = *(const f4*)(base + (size_t)i * D + d);
                acc += ws[i] * v;
            }
            *(f4*)(ob + d) = acc;
        } else {
            for (long long dd = d; dd < D; ++dd) {
                float a = 0.f;
                #pragma unroll
                for (int i = 0; i < NG; ++i) a += ws[i] * base[(size_t)i * D + dd];
                ob[dd] = a;
            }
        }
    }
}

extern "C" void kernel_launch(void* const* d_in, const int* in_sizes, int n_in,
                              void* d_out, int out_size, void* d_ws, size_t ws_size,
                              hipStream_t stream) {
    const float* g = (const float*)d_in[0];
    float* out = (float*)d_out;

    const long long total = (long long)in_sizes[0];       // P * G * D
    const long long D = total / (NP * NG);

    float* gram = (float*)d_ws;                           // NP * 256 floats
    float* w    = gram + NP * NG * NG;                    // NP * 16 floats

    // stage 0: zero the Gram accumulators (atomics re-accumulate each call)
    {
        int n = NP * NG * NG;
        zero_ws_kernel<<<(n + 255) / 256, 256, 0, stream>>>(gram, n);
    }

    // stage 1: Gram via WMMA. 512 blocks x 8 waves = 4096 waves per parameter
    // (~32K waves total -> enough outstanding 128b loads to cover HBM latency).
    {
        const int blocksX = 512;
        const long long wavesPerParam = (long long)blocksX * 8;
        long long chunk = (D + wavesPerParam - 1) / wavesPerParam;
        chunk = ((chunk + 15) / 16) * 16;                 // multiple of 16 cols
        gram_wmma_kernel<<<dim3(blocksX, NP), 256, 0, stream>>>(g, gram, D, chunk);
    }

    // stage 2: tiny per-parameter weight solve
    weights_kernel<<<NP, 32, 0, stream>>>(gram, w);

    // stage 3: weighted row sum, one float4 per thread
    {
        long long nq = (D + 3) / 4;
        long long gx = (nq + 255) / 256;
        if (gx < 1) gx = 1;
        if (gx > 65535) gx = 65535;                       // grid-stride covers rest
        wsum_kernel<<<dim3((unsigned)gx, NP), 256, 0, stream>>>(g, w, out, D);
    }
}